// InnerProduct_541165879452
// MI455X (gfx1250) — compile-verified
//
#include <hip/hip_runtime.h>

typedef __attribute__((ext_vector_type(2))) float v2f;
typedef __attribute__((ext_vector_type(4))) float v4f;
typedef __attribute__((ext_vector_type(8))) float v8f;

#define F_DIM 64
#define E_DIM 16
#define P_DIM 10

#define ROWS_PER_WAVE  16
#define WAVES_PER_BLOCK 8
#define ROWS_PER_BLOCK (ROWS_PER_WAVE * WAVES_PER_BLOCK)   // 128
#define XSQ_PITCH 68   // 64 + 4 pad -> stage-B lane L hits banks {4L, 4L+1}: conflict-free

__global__ __launch_bounds__(256)
void lp_wmma_kernel(const float* __restrict__ x,
                    const float* __restrict__ w,
                    float* __restrict__ out, int R) {
    __shared__ float xsq[WAVES_PER_BLOCK * ROWS_PER_WAVE * XSQ_PITCH];

    const int tid  = threadIdx.x;
    const int wv   = tid >> 5;          // wave id in block (wave32)
    const int lane = tid & 31;
    const int rb   = blockIdx.x * ROWS_PER_BLOCK;
    const int wrb  = rb + wv * ROWS_PER_WAVE;   // this wave's 16-row tile base
    const bool full_tile = (wrb + ROWS_PER_WAVE) <= R;  // always true when R % 128 == 0

    // ---------- preload B fragments: w_sq in V_WMMA_F32_16X16X4_F32 B layout ----------
    // B is 4x16 (KxN): lanes 0-15 hold N=lane, K={k0,k0+1}; lanes 16-31 hold K={k0+2,k0+3}.
    const int p    = lane & 15;
    const int koff = (lane >> 4) << 1;  // 0 or 2
    v2f bfrag[16];
    #pragma unroll
    for (int s = 0; s < 16; ++s) {
        int k = s * 4 + koff;
        float bx = 0.0f, by = 0.0f;
        if (p < P_DIM) {
            float w0 = w[p * F_DIM + k];
            float w1 = w[p * F_DIM + k + 1];
            bx = w0 * w0;
            by = w1 * w1;
        }
        bfrag[s].x = bx;
        bfrag[s].y = by;
    }

    // ---------- stage A: x_sq[r][f] = sum_e x[r][f][e]^2 for this wave's 16 rows ----------
    // 16 rows * 64 f = 1024 (r,f) pairs; each lane fully reduces one pair per iteration
    // (4 x global_load_b128 over the 16 contiguous E-values -> register FMA tree).
    float* myxsq = &xsq[wv * ROWS_PER_WAVE * XSQ_PITCH];
    const float* xb = x + (size_t)wrb * (F_DIM * E_DIM);
    if (full_tile) {
        #pragma unroll 4
        for (int it = 0; it < 32; ++it) {
            int pair = it * 32 + lane;      // 0..1023
            int rl   = pair >> 6;           // local row 0..15
            int f    = pair & 63;
            const float* src = xb + (size_t)pair * E_DIM;
            float s = 0.0f;
            #pragma unroll
            for (int j = 0; j < 4; ++j) {
                v4f v = *(const v4f*)(src + j * 4);
                s = fmaf(v.x, v.x, s);
                s = fmaf(v.y, v.y, s);
                s = fmaf(v.z, v.z, s);
                s = fmaf(v.w, v.w, s);
            }
            myxsq[rl * XSQ_PITCH + f] = s;
        }
    } else {
        for (int it = 0; it < 32; ++it) {
            int pair = it * 32 + lane;
            int rl   = pair >> 6;
            int f    = pair & 63;
            float s = 0.0f;
            if (wrb + rl < R) {
                const float* src = xb + (size_t)pair * E_DIM;
                #pragma unroll
                for (int j = 0; j < 4; ++j) {
                    v4f v = *(const v4f*)(src + j * 4);
                    s = fmaf(v.x, v.x, s);
                    s = fmaf(v.y, v.y, s);
                    s = fmaf(v.z, v.z, s);
                    s = fmaf(v.w, v.w, s);
                }
            }
            myxsq[rl * XSQ_PITCH + f] = s;
        }
    }

    __syncthreads();

    // ---------- stage B: lp tile (16x16) = A(16x64) x B(64x16), 16 x WMMA f32 16x16x4 ----------
    // A layout: lanes 0-15: M=lane, VGPR0=K0,VGPR1=K1; lanes 16-31: M=lane-16, K2,K3.
    const int m = lane & 15;
    v8f c = {};
    #pragma unroll
    for (int s = 0; s < 16; ++s) {
        int k = s * 4 + koff;
        v2f a = *(const v2f*)&myxsq[m * XSQ_PITCH + k];   // ds_load_2addr_b64, conflict-free
        c = __builtin_amdgcn_wmma_f32_16x16x4_f32(
                /*neg_a=*/false, a, /*neg_b=*/false, bfrag[s],
                /*c_mod=*/(short)0, c, /*reuse_a=*/false, /*reuse_b=*/false);
    }

    // ---------- store D: row = i + 8*(lane>=16), col = lane&15 (only p < 10 valid) ----------
    // Fast path: one 64-bit base + 8 global_store_b32 with immediate offsets (stride 40 B).
    const int mb = (lane >> 4) << 3;
    if (p < P_DIM) {
        float* dst = out + (size_t)(wrb + mb) * P_DIM + p;
        if (full_tile) {
            #pragma unroll
            for (int i = 0; i < 8; ++i)
                dst[i * P_DIM] = c[i];
        } else {
            #pragma unroll
            for (int i = 0; i < 8; ++i)
                if (wrb + mb + i < R) dst[i * P_DIM] = c[i];
        }
    }
}

extern "C" void kernel_launch(void* const* d_in, const int* in_sizes, int n_in,
                              void* d_out, int out_size, void* d_ws, size_t ws_size,
                              hipStream_t stream) {
    const float* x = (const float*)d_in[0];   // [R, 64, 16] f32
    const float* w = (const float*)d_in[1];   // [10, 64]   f32
    float* out = (float*)d_out;               // [R, 10]    f32

    int R = in_sizes[0] / (F_DIM * E_DIM);    // 32768 for the reference shapes
    int blocks = (R + ROWS_PER_BLOCK - 1) / ROWS_PER_BLOCK;
    lp_wmma_kernel<<<blocks, 256, 0, stream>>>(x, w, out, R);
}